// Attention_39170101739612
// MI455X (gfx1250) — compile-verified
//
#include <hip/hip_runtime.h>
#include <hip/hip_bf16.h>
#include <math.h>

#define K_MASK_C (-2.3819763e+38f)

typedef float v2f __attribute__((ext_vector_type(2)));
typedef float v8f __attribute__((ext_vector_type(8)));

// Problem dims (fixed by reference)
#define BB 8
#define SS 4096
#define NN 32
#define HH 128
#define DD 4096

#define KC  512              // K-chunk staged in LDS
#define PAD 4                // LDS row pad (floats) -> rows hit distinct banks

#define KSPLIT1 2            // split-K for QKV projection  (K=4096 -> 2x2048)
#define KSPLIT4 4            // split-K for output projection (K=4096 -> 4x1024)

// ---------------------------------------------------------------------------
// Zero rows 8..15 of the LDS A tile once (A logically has 16 rows, 8 real).
// ---------------------------------------------------------------------------
__device__ inline void zero_hi_rows(float (*As)[KC + PAD])
{
    float* z = &As[8][0];                       // rows 8..15 contiguous
    for (int i = threadIdx.x; i < 8 * (KC + PAD); i += 256)
        z[i] = 0.0f;
}

// ---------------------------------------------------------------------------
// Cooperative stage of A chunk (rows 0..7 x KC floats) into LDS, 256 threads.
// ---------------------------------------------------------------------------
__device__ inline void stage_A(float (*As)[KC + PAD],
                               const float* __restrict__ A, int lda, int kc0)
{
    const int tid = threadIdx.x;
    constexpr int vecs = 8 * KC / 4;            // float4 count
    #pragma unroll
    for (int i = tid; i < vecs; i += 256) {
        const int row = i / (KC / 4);
        const int c4  = (i % (KC / 4)) * 4;
        const float4 v = *(const float4*)(A + (size_t)row * lda + kc0 + c4);
        *(float4*)&As[row][c4] = v;
    }
}

// ---------------------------------------------------------------------------
// One K-chunk of a 16x16 tile: acc += A_lds[16 x KC] * Bt[KC x 16].
// Pure ds_load + global_load + wmma: no per-iteration masking VALU.
// ---------------------------------------------------------------------------
__device__ inline void wmma_chunk(const float (*As)[KC + PAD],
                                  const float* __restrict__ Bt, int ldb,
                                  v8f& acc)
{
    const int lane = threadIdx.x & 31;
    const int half = lane >> 4;          // 0: lanes 0-15, 1: lanes 16-31
    const int l    = lane & 15;          // N index (B/C) and M row index (A)

    for (int k = 0; k < KC; k += 4) {
        const int ka = k + half * 2;     // K pair handled by this half-wave
        v2f a, b;
        a.x = As[l][ka];
        a.y = As[l][ka + 1];
        b.x = Bt[(size_t)ka * ldb + l];
        b.y = Bt[(size_t)(ka + 1) * ldb + l];
        acc = __builtin_amdgcn_wmma_f32_16x16x4_f32(false, a, false, b,
                                                    (short)0, acc, false, false);
    }
}

// ---------------------------------------------------------------------------
// K1: fused QKV projection (split-K x2, deterministic partial buffers).
// part[kp][sel][b][n*128+h] = x[b, kp*2048:(kp+1)*2048] . qkv_w[sel,n,slice,h]
// Block = 8 waves = one 128-column stripe = one (sel,n) weight block.
// ---------------------------------------------------------------------------
__global__ void __launch_bounds__(256) qkv_proj_kernel(
    const float* __restrict__ x,       // [8][4096]
    const float* __restrict__ qkv_w,   // [3][32][4096][128]
    float* __restrict__ qkv_part)      // [KSPLIT1][3][8][4096]
{
    __shared__ __align__(16) float As[16][KC + PAD];

    const int kp       = blockIdx.x & (KSPLIT1 - 1);
    const int tile_grp = blockIdx.x / KSPLIT1;        // 0..95
    const int wave = threadIdx.x >> 5;
    const int tile = tile_grp * 8 + wave;             // 0..767
    const int col0 = tile * 16;
    const int blk  = col0 >> 7;                       // flattened (sel*32 + n)
    const int h0   = col0 & 127;
    const int k0   = kp * (DD / KSPLIT1);
    const float* Bt = qkv_w + (size_t)blk * DD * HH + h0;

    zero_hi_rows(As);

    v8f acc = {0.f, 0.f, 0.f, 0.f, 0.f, 0.f, 0.f, 0.f};
    for (int kc = 0; kc < DD / KSPLIT1; kc += KC) {
        __syncthreads();
        stage_A(As, x, DD, k0 + kc);
        __syncthreads();
        wmma_chunk(As, Bt + (size_t)(k0 + kc) * HH, HH, acc);
    }

    const int lane = threadIdx.x & 31;
    if (lane < 16) {
        const int colg   = col0 + lane;         // C element (M=r, N=lane)
        const int sel    = colg >> 12;          // / (32*128)
        const int within = colg & 4095;         // n*128 + h
        float* dst = qkv_part + (size_t)kp * 3 * BB * (NN * HH);
        #pragma unroll
        for (int r = 0; r < 8; ++r)             // r == batch b
            dst[(size_t)(sel * BB + r) * (NN * HH) + within] = acc[r];
    }
}

// ---------------------------------------------------------------------------
// K2: combine split-K partials (fixed order), RoPE(q)*H^-0.5 -> qbuf,
// RoPE(k) + v scattered into the output caches.
// ---------------------------------------------------------------------------
__global__ void rope_scatter_kernel(
    const float* __restrict__ qkv_part, // [KSPLIT1][3][8][32*128]
    float* __restrict__ qbuf,           // [8][32*128]
    float* __restrict__ out_k,          // [8][4096][32][128]
    float* __restrict__ out_v,
    const int* __restrict__ seg_pos,
    const int* __restrict__ tstep)
{
    const int b = blockIdx.x >> 5;
    const int n = blockIdx.x & 31;
    const int t = threadIdx.x;        // 0..127
    const int pos = seg_pos[0];
    const int ts  = tstep[0];

    const size_t PSTRIDE = (size_t)3 * BB * (NN * HH);
    const float* p0 = qkv_part;
    const float* p1 = qkv_part + PSTRIDE;

    const size_t qo = (size_t)b * (NN * HH) + n * HH;
    const size_t ko = (size_t)(BB + b) * (NN * HH) + n * HH;
    const size_t vo = (size_t)(2 * BB + b) * (NN * HH) + n * HH;

    float* okrow = out_k + (((size_t)b * SS + ts) * NN + n) * HH;
    float* ovrow = out_v + (((size_t)b * SS + ts) * NN + n) * HH;

    ovrow[t] = p0[vo + t] + p1[vo + t];          // fixed summation order

    if (t < HH / 2) {
        const float frac      = (2.0f * (float)t) / (float)HH;
        const float timescale = __expf(frac * 9.210340371976184f); // 10000^frac
        const float sinus     = (float)pos / timescale;
        float sn, cs;
        __sincosf(sinus, &sn, &cs);

        const float kf = p0[ko + t] + p1[ko + t];
        const float ks = p0[ko + t + HH / 2] + p1[ko + t + HH / 2];
        okrow[t]          = kf * cs - ks * sn;
        okrow[t + HH / 2] = ks * cs + kf * sn;

        const float sc = 0.0883883476483184f;    // 128^-0.5
        const float qf = p0[qo + t] + p1[qo + t];
        const float qs = p0[qo + t + HH / 2] + p1[qo + t + HH / 2];
        qbuf[qo + t]          = (qf * cs - qs * sn) * sc;
        qbuf[qo + t + HH / 2] = (qs * cs + qf * sn) * sc;
    }
}

// ---------------------------------------------------------------------------
// K3: flash-decode attention. One block per (b,n); 8 waves stream S-chunks with
// online softmax; masked rows contribute exactly 0 -> skip their K/V loads.
// ---------------------------------------------------------------------------
__global__ void __launch_bounds__(256) attn_decode_kernel(
    const float* __restrict__ qbuf,     // [8][32*128] (scaled, rotated)
    const float* __restrict__ cache_k,  // updated caches in d_out
    const float* __restrict__ cache_v,
    const float* __restrict__ mask,     // [8][4096]
    float* __restrict__ enc)            // [8][32*128]
{
    const int b    = blockIdx.x >> 5;
    const int n    = blockIdx.x & 31;
    const int wave = threadIdx.x >> 5;  // 0..7
    const int lane = threadIdx.x & 31;

    const float4 qv = *(const float4*)(qbuf + (size_t)b * (NN * HH) + n * HH + lane * 4);
    const float* mrow = mask + (size_t)b * SS;

    float  m    = -3.0e38f;
    float  ssum = 0.f;
    float4 accv = {0.f, 0.f, 0.f, 0.f};

    const int s0 = wave * (SS / 8);
    for (int s = s0; s < s0 + SS / 8; ++s) {
        if (mrow[s] < K_MASK_C * 0.5f) continue;   // prob would be exactly 0
        const size_t row = (((size_t)b * SS + s) * NN + n) * HH + lane * 4;
        const float4 kv = *(const float4*)(cache_k + row);
        float d = qv.x * kv.x + qv.y * kv.y + qv.z * kv.z + qv.w * kv.w;
        #pragma unroll
        for (int off = 16; off >= 1; off >>= 1)
            d += __shfl_xor(d, off, 32);
        const float mn    = fmaxf(m, d);
        const float scale = __expf(m - mn);
        const float p     = __expf(d - mn);
        ssum = ssum * scale + p;
        const float4 vv = *(const float4*)(cache_v + row);
        accv.x = accv.x * scale + p * vv.x;
        accv.y = accv.y * scale + p * vv.y;
        accv.z = accv.z * scale + p * vv.z;
        accv.w = accv.w * scale + p * vv.w;
        m = mn;
    }

    __shared__ float sm[8], ssm[8];
    __shared__ __align__(16) float sa[8][HH];
    if (lane == 0) { sm[wave] = m; ssm[wave] = ssum; }
    ((float4*)sa[wave])[lane] = accv;
    __syncthreads();

    if (threadIdx.x < HH) {
        const int h = threadIdx.x;
        float M = sm[0];
        #pragma unroll
        for (int w = 1; w < 8; ++w) M = fmaxf(M, sm[w]);
        float denom = 0.f, val = 0.f;
        #pragma unroll
        for (int w = 0; w < 8; ++w) {
            const float e = __expf(sm[w] - M);
            denom += e * ssm[w];
            val   += e * sa[w][h];
        }
        enc[(size_t)b * (NN * HH) + n * HH + h] = val / denom;
    }
}

// ---------------------------------------------------------------------------
// K4: output projection (split-K x4). opart[kp][b][d] = partial dot.
// Block = 8 waves = one 128-column stripe of the 4096 output columns.
// ---------------------------------------------------------------------------
__global__ void __launch_bounds__(256) out_proj_kernel(
    const float* __restrict__ enc,   // [8][4096]
    const float* __restrict__ w,     // [32*128][4096] row-major in (n*128+h)
    float* __restrict__ opart)       // [KSPLIT4][8][4096]
{
    __shared__ __align__(16) float As[16][KC + PAD];

    const int kp       = blockIdx.x & (KSPLIT4 - 1);
    const int tile_grp = blockIdx.x / KSPLIT4;        // 0..31
    const int wave = threadIdx.x >> 5;
    const int tile = tile_grp * 8 + wave;             // 0..255
    const int d0   = tile * 16;
    const int k0   = kp * (NN * HH / KSPLIT4);

    zero_hi_rows(As);

    v8f acc = {0.f, 0.f, 0.f, 0.f, 0.f, 0.f, 0.f, 0.f};
    for (int kc = 0; kc < NN * HH / KSPLIT4; kc += KC) {
        __syncthreads();
        stage_A(As, enc, NN * HH, k0 + kc);
        __syncthreads();
        wmma_chunk(As, w + (size_t)(k0 + kc) * DD + d0, DD, acc);
    }

    const int lane = threadIdx.x & 31;
    if (lane < 16) {
        float* dst = opart + (size_t)kp * BB * DD;
        #pragma unroll
        for (int r = 0; r < 8; ++r)             // r == batch b
            dst[(size_t)r * DD + d0 + lane] = acc[r];
    }
}

// ---------------------------------------------------------------------------
// K5: fixed-order reduction of the split-K output partials.
// ---------------------------------------------------------------------------
__global__ void reduce_out_kernel(const float* __restrict__ opart,
                                  float* __restrict__ out)
{
    const int i = blockIdx.x * blockDim.x + threadIdx.x;   // 0..32767
    const size_t P = (size_t)BB * DD;
    float v = opart[i];                                    // fixed order
    v += opart[P + i];
    v += opart[2 * P + i];
    v += opart[3 * P + i];
    out[i] = v;
}

// ---------------------------------------------------------------------------
extern "C" void kernel_launch(void* const* d_in, const int* in_sizes, int n_in,
                              void* d_out, int out_size, void* d_ws, size_t ws_size,
                              hipStream_t stream)
{
    (void)in_sizes; (void)n_in; (void)out_size; (void)ws_size;

    const float* x       = (const float*)d_in[0];
    const float* cache_k = (const float*)d_in[1];
    const float* cache_v = (const float*)d_in[2];
    const float* mask    = (const float*)d_in[3];
    const float* qkv_w   = (const float*)d_in[4];
    const float* avec_w  = (const float*)d_in[5];
    const int*   seg_pos = (const int*)d_in[6];
    const int*   tstep   = (const int*)d_in[7];

    const size_t CE = (size_t)BB * SS * NN * HH;   // 134217728 elements
    float* out_k = (float*)d_out;
    float* out_v = out_k + CE;
    float* out_a = out_v + CE;

    // Workspace layout (floats)
    float* qkv_part = (float*)d_ws;                               // 2 * 98304
    float* qbuf     = qkv_part + (size_t)KSPLIT1 * 3 * BB * NN * HH;  // 32768
    float* enc_ws   = qbuf + (size_t)BB * NN * HH;                    // 32768
    float* opart    = enc_ws + (size_t)BB * NN * HH;              // 4 * 32768

    // Bulk cache copy into output (dominant traffic; async d2d on stream).
    hipMemcpyAsync(out_k, cache_k, CE * sizeof(float), hipMemcpyDeviceToDevice, stream);
    hipMemcpyAsync(out_v, cache_v, CE * sizeof(float), hipMemcpyDeviceToDevice, stream);

    // QKV projection: 192 blocks x 8 WMMA waves (split-K x2)
    qkv_proj_kernel<<<96 * KSPLIT1, 256, 0, stream>>>(x, qkv_w, qkv_part);
    // Combine partials + RoPE + scatter new K/V rows into the output caches
    rope_scatter_kernel<<<BB * NN, 128, 0, stream>>>(qkv_part, qbuf, out_k, out_v,
                                                     seg_pos, tstep);
    // Flash-decode attention over updated caches
    attn_decode_kernel<<<BB * NN, 256, 0, stream>>>(qbuf, out_k, out_v, mask, enc_ws);
    // Output projection: 128 blocks x 8 WMMA waves (split-K x4)
    out_proj_kernel<<<32 * KSPLIT4, 256, 0, stream>>>(enc_ws, avec_w, opart);
    // Deterministic fixed-order reduction of output partials
    reduce_out_kernel<<<(BB * DD) / 256, 256, 0, stream>>>(opart, out_a);
}